// MultiLabelAngleProtoCELoss_75015898792460
// MI455X (gfx1250) — compile-verified
//
#include <hip/hip_runtime.h>
#include <hip/hip_bf16.h>
#include <math.h>

typedef __attribute__((ext_vector_type(2))) float v2f;
typedef __attribute__((ext_vector_type(8))) float v8f;

#define BROWS 16384
#define DDIM  1024
#define NLAB  64
#define CLAB  65          // 64 labels + "no-label" column
#define CPAD  80          // padded to multiple of 16 for WMMA
#define EPSN  1e-8f
#define NEGBIG (-1e30f)

// ---------------- K0: zero counts ----------------
__global__ __launch_bounds__(96) void zero_counts_kernel(float* counts) {
    if (threadIdx.x < CPAD) counts[threadIdx.x] = 0.0f;
}

// ---------------- K1: build lab matrices + counts ----------------
__global__ __launch_bounds__(256) void build_lab_kernel(const int* __restrict__ label,
                                                        float* __restrict__ labf,
                                                        float* __restrict__ labT,
                                                        float* __restrict__ counts) {
    __shared__ float lc[CPAD];
    if (threadIdx.x < CPAD) lc[threadIdx.x] = 0.0f;
    __syncthreads();

    const int row = blockIdx.x * 256 + threadIdx.x;   // 64 blocks * 256 = 16384
    int s = 0;
    for (int j = 0; j < NLAB; ++j) {
        int v = label[(size_t)row * NLAB + j];        // 0 or 1
        float f = (float)v;
        labf[(size_t)row * CPAD + j] = f;
        labT[(size_t)j * BROWS + row] = f;
        if (v) { atomicAdd(&lc[j], 1.0f); s += v; }
    }
    float extra = (s == 0) ? 1.0f : 0.0f;
    labf[(size_t)row * CPAD + NLAB] = extra;
    labT[(size_t)NLAB * BROWS + row] = extra;
    if (s == 0) atomicAdd(&lc[NLAB], 1.0f);
    for (int j = CLAB; j < CPAD; ++j) {
        labf[(size_t)row * CPAD + j] = 0.0f;
        labT[(size_t)j * BROWS + row] = 0.0f;
    }
    __syncthreads();
    // counts are sums of 1.0f: exact in f32 regardless of order -> deterministic
    if (threadIdx.x < CPAD) atomicAdd(&counts[threadIdx.x], lc[threadIdx.x]);
}

// ---------------- K1b: per-row L2 norms of x ----------------
__global__ __launch_bounds__(256) void row_norms_kernel(const float* __restrict__ x,
                                                        float* __restrict__ xn) {
    const int wv = threadIdx.x >> 5, lane = threadIdx.x & 31;
    const int row = blockIdx.x * 8 + wv;              // 2048 blocks * 8 waves
    const float* p = x + (size_t)row * DDIM;
    float s = 0.0f;
    for (int k = lane; k < DDIM; k += 32) { float v = p[k]; s += v * v; }
    #pragma unroll
    for (int off = 16; off > 0; off >>= 1) s += __shfl_xor(s, off);
    if (lane == 0) xn[row] = fmaxf(sqrtf(s), EPSN);
}

// ---------------- K2: anchor = lab^T @ x / counts  (WMMA f32 16x16x4) ----------------
// M = CPAD(80) -> 5 tiles, N = DDIM(1024) -> 64 tiles, K = BROWS(16384) split over 8 waves.
__global__ __launch_bounds__(256) void anchor_gemm_kernel(const float* __restrict__ labT,
                                                          const float* __restrict__ x,
                                                          const float* __restrict__ counts,
                                                          float* __restrict__ anchor) {
    const int tile  = blockIdx.x;         // 0..319
    const int mtile = tile >> 6;          // /64 -> 0..4
    const int ntile = tile & 63;          // 0..63
    const int lane  = threadIdx.x & 31;
    const int wv    = threadIdx.x >> 5;   // 0..7 K-splits
    const int mn    = lane & 15;          // A: M index, B: N index
    const int koff  = (lane >> 4) << 1;   // K pair: lanes 0-15 -> K{0,1}, 16-31 -> K{2,3}

    const float* aptr = labT + (size_t)(mtile * 16 + mn) * BROWS;  // A row (= label col)
    const int bcol = ntile * 16 + mn;

    v8f acc = {};
    const int kseg = BROWS / 8;
    const int k0 = wv * kseg;
    for (int k = k0; k < k0 + kseg; k += 4) {
        v2f a = *(const v2f*)(aptr + k + koff);                       // b64 load
        v2f b;
        b.x = x[(size_t)(k + koff)     * DDIM + bcol];
        b.y = x[(size_t)(k + koff + 1) * DDIM + bcol];
        acc = __builtin_amdgcn_wmma_f32_16x16x4_f32(
                  false, a, false, b, (short)0, acc, false, false);
    }

    // cross-wave K reduction through LDS
    __shared__ float red[8][256];
    #pragma unroll
    for (int j = 0; j < 8; ++j) red[wv][lane * 8 + j] = acc[j];
    __syncthreads();

    const int t  = threadIdx.x;           // one thread per tile element
    const int rl = t >> 3, rj = t & 7;    // source lane / source VGPR j
    float s = 0.0f;
    #pragma unroll
    for (int w = 0; w < 8; ++w) s += red[w][rl * 8 + rj];

    // C/D layout: M = j + 8*(lane/16), N = lane%16
    const int mrow = rj + 8 * (rl >> 4);
    const int ncol = rl & 15;
    const int c = mtile * 16 + mrow;
    const int d = ntile * 16 + ncol;
    const float cnt = (c < CLAB) ? counts[c] : 0.0f;
    anchor[(size_t)c * DDIM + d] = s / fmaxf(cnt, 1.0f);
}

// ---------------- K2b: anchor norms ----------------
__global__ __launch_bounds__(32) void anchor_norms_kernel(const float* __restrict__ anchor,
                                                          float* __restrict__ an) {
    const int c = blockIdx.x, lane = threadIdx.x;
    const float* p = anchor + (size_t)c * DDIM;
    float s = 0.0f;
    for (int k = lane; k < DDIM; k += 32) { float v = p[k]; s += v * v; }
    #pragma unroll
    for (int off = 16; off > 0; off >>= 1) s += __shfl_xor(s, off);
    if (lane == 0) an[c] = fmaxf(sqrtf(s), EPSN);
}

// ---------------- K3: cos GEMM (WMMA) fused with masked log-softmax loss ----------------
// One block = 16 sample rows; 5 waves, wave wv computes 16x16 dot tile for cols [wv*16, wv*16+16).
__global__ __launch_bounds__(160) void cos_loss_kernel(const float* __restrict__ x,
                                                       const float* __restrict__ anchor,
                                                       const float* __restrict__ xn,
                                                       const float* __restrict__ an,
                                                       const float* __restrict__ counts,
                                                       const float* __restrict__ labf,
                                                       const float* __restrict__ wp,
                                                       const float* __restrict__ bp,
                                                       float* __restrict__ blockloss) {
    const int mtile = blockIdx.x;         // 0..1023
    const int wv    = threadIdx.x >> 5;   // 0..4 == ntile
    const int lane  = threadIdx.x & 31;
    const int mn    = lane & 15;
    const int koff  = (lane >> 4) << 1;

    const float* aptr = x      + (size_t)(mtile * 16 + mn) * DDIM;   // A: x rows
    const float* bptr = anchor + (size_t)(wv    * 16 + mn) * DDIM;   // B: anchor^T cols

    v8f acc = {};
    for (int k = 0; k < DDIM; k += 4) {
        v2f a = *(const v2f*)(aptr + k + koff);   // b64 contiguous
        v2f b = *(const v2f*)(bptr + k + koff);   // b64 contiguous
        acc = __builtin_amdgcn_wmma_f32_16x16x4_f32(
                  false, a, false, b, (short)0, acc, false, false);
    }

    __shared__ float sdot[16 * CPAD];
    #pragma unroll
    for (int j = 0; j < 8; ++j) {
        const int mrow = j + 8 * (lane >> 4);
        const int ncol = lane & 15;
        sdot[mrow * CPAD + wv * 16 + ncol] = acc[j];
    }
    __syncthreads();

    __shared__ float rowloss[16];
    if (threadIdx.x < 16) {
        const int r = threadIdx.x;
        const int row = mtile * 16 + r;
        const float invx = 1.0f / xn[row];
        const float W = wp[0], B = bp[0];
        float mx = NEGBIG;
        for (int c = 0; c < CLAB; ++c) {
            float lg;
            if (counts[c] > 0.0f) lg = (sdot[r * CPAD + c] * invx / an[c]) * W + B;
            else                  lg = NEGBIG;           // masked column
            sdot[r * CPAD + c] = lg;
            mx = fmaxf(mx, lg);
        }
        float se = 0.0f;
        for (int c = 0; c < CLAB; ++c) se += expf(sdot[r * CPAD + c] - mx);
        const float lse = logf(se) + mx;
        float loss = 0.0f;
        for (int c = 0; c < CLAB; ++c) {
            const float lv = labf[(size_t)row * CPAD + c];
            if (lv > 0.0f) loss += lv * (lse - sdot[r * CPAD + c]);   // -logp, pair-weighted
        }
        rowloss[r] = loss;
    }
    __syncthreads();
    if (threadIdx.x == 0) {
        float s = 0.0f;
        #pragma unroll
        for (int i = 0; i < 16; ++i) s += rowloss[i];
        blockloss[mtile] = s;
    }
}

// ---------------- K4: final reduction ----------------
__global__ __launch_bounds__(256) void final_reduce_kernel(const float* __restrict__ blockloss,
                                                           const float* __restrict__ counts,
                                                           float* __restrict__ out) {
    __shared__ float sm[256];
    float s = 0.0f;
    for (int i = threadIdx.x; i < 1024; i += 256) s += blockloss[i];
    sm[threadIdx.x] = s;
    __syncthreads();
    for (int off = 128; off > 0; off >>= 1) {
        if (threadIdx.x < off) sm[threadIdx.x] += sm[threadIdx.x + off];
        __syncthreads();
    }
    if (threadIdx.x == 0) {
        float np = 0.0f;
        for (int c = 0; c < CLAB; ++c) np += counts[c];   // n_pairs (exact integers)
        out[0] = sm[0] / np;
    }
}

extern "C" void kernel_launch(void* const* d_in, const int* in_sizes, int n_in,
                              void* d_out, int out_size, void* d_ws, size_t ws_size,
                              hipStream_t stream) {
    (void)in_sizes; (void)n_in; (void)out_size; (void)ws_size;
    const float* x     = (const float*)d_in[0];
    const int*   label = (const int*)  d_in[1];
    const float* wpar  = (const float*)d_in[2];
    const float* bpar  = (const float*)d_in[3];
    float*       out   = (float*)d_out;

    // workspace layout (floats)
    float* ws      = (float*)d_ws;
    float* labT    = ws;                                   // 80 * 16384
    float* labf    = labT   + (size_t)CPAD * BROWS;        // 16384 * 80
    float* counts  = labf   + (size_t)BROWS * CPAD;        // 80
    float* xn      = counts + CPAD;                        // 16384
    float* anchor  = xn     + BROWS;                       // 80 * 1024
    float* an      = anchor + (size_t)CPAD * DDIM;         // 80
    float* bloss   = an     + CPAD;                        // 1024

    zero_counts_kernel<<<1, 96, 0, stream>>>(counts);
    build_lab_kernel<<<BROWS / 256, 256, 0, stream>>>(label, labf, labT, counts);
    row_norms_kernel<<<BROWS / 8, 256, 0, stream>>>(x, xn);
    anchor_gemm_kernel<<<(CPAD / 16) * (DDIM / 16), 256, 0, stream>>>(labT, x, counts, anchor);
    anchor_norms_kernel<<<CPAD, 32, 0, stream>>>(anchor, an);
    cos_loss_kernel<<<BROWS / 16, 160, 0, stream>>>(x, anchor, xn, an, counts, labf,
                                                    wpar, bpar, bloss);
    final_reduce_kernel<<<1, 256, 0, stream>>>(bloss, counts, out);
}